// GNet_12867722019170
// MI455X (gfx1250) — compile-verified
//
#include <hip/hip_runtime.h>

typedef __attribute__((ext_vector_type(4))) float v4f;
typedef int v4i __attribute__((vector_size(16)));   // matches builtin param type

// pointer-to-AS1-v4i / pointer-to-AS3-v4i
typedef __attribute__((address_space(1))) v4i* gas_v4i_ptr;
typedef __attribute__((address_space(3))) v4i* las_v4i_ptr;

#define NB_IOU 0.2f

// ---------------------------------------------------------------------------
// CDNA5 async copy: global -> LDS, 128-bit, tracked by ASYNCcnt.
// ---------------------------------------------------------------------------
__device__ __forceinline__ void async_copy_b128_to_lds(v4f* lds_dst, const v4f* g_src) {
#if __has_builtin(__builtin_amdgcn_global_load_async_to_lds_b128)
    __builtin_amdgcn_global_load_async_to_lds_b128(
        (gas_v4i_ptr)(void*)g_src,
        (las_v4i_ptr)(void*)lds_dst,
        /*imm offset*/0, /*cpol*/0);
#else
    unsigned lds_off = (unsigned)(unsigned long long)(void*)lds_dst; // low 32b = LDS byte addr
    asm volatile("global_load_async_to_lds_b128 %0, %1, off"
                 :: "v"(lds_off), "v"(g_src) : "memory");
#endif
}

__device__ __forceinline__ void wait_asynccnt0() {
#if __has_builtin(__builtin_amdgcn_s_wait_asynccnt)
    __builtin_amdgcn_s_wait_asynccnt(0);
#else
    asm volatile("s_wait_asynccnt 0" ::: "memory");
#endif
}

// ---------------------------------------------------------------------------
// Pairwise IoU tile kernel.
// Block = 256 threads (8 wave32). Each thread computes a 4x4 tile:
//   rows  4*rg .. 4*rg+3   (rg = blockIdx.y*4 + tid/64)
//   cols  4*cg .. 4*cg+3   (cg = blockIdx.x*64 + tid%64)
// Block stages its 256 column boxes (4 KB) into LDS via the async copy engine.
// Consecutive lanes -> consecutive column groups -> coalesced b128 NT stores.
// ---------------------------------------------------------------------------
template <bool WRITE_MASK>
__global__ __launch_bounds__(256)
void iou_tile_kernel(const v4f* __restrict__ rows, int nrows,
                     const v4f* __restrict__ cols, int ncols,
                     float* __restrict__ out_iou,
                     float* __restrict__ out_mask) {
    __shared__ v4f sCols[256];

    const int tid       = threadIdx.x;
    const int colGroups = ncols >> 2;            // groups of 4 columns
    const int colBase   = blockIdx.x * 256;      // first column staged by this block

    // ---- stage column boxes into LDS (async copy engine) ----
    const int remain = ncols - colBase;
    const int nStage = remain < 256 ? remain : 256;
    if (tid < nStage) {
        async_copy_b128_to_lds(&sCols[tid], &cols[colBase + tid]);
    }
    wait_asynccnt0();      // per-wave ASYNCcnt drain
    __syncthreads();       // cross-wave visibility of LDS

    const int lc = tid & 63;                     // local column group
    const int cg = blockIdx.x * 64 + lc;         // global column group
    const int rg = blockIdx.y * 4 + (tid >> 6);  // global row group
    if (cg >= colGroups || (rg << 2) >= nrows) return;

    // ---- load 4 column boxes from LDS, precompute areas ----
    v4f   cb[4];
    float carea[4];
#pragma unroll
    for (int k = 0; k < 4; ++k) {
        cb[k]    = sCols[(lc << 2) + k];
        carea[k] = (cb[k].z - cb[k].x) * (cb[k].w - cb[k].y);
    }

#pragma unroll
    for (int r = 0; r < 4; ++r) {
        const int row = (rg << 2) + r;
        const v4f rb  = rows[row];               // wave-uniform broadcast load
        const float rarea = (rb.z - rb.x) * (rb.w - rb.y);

        v4f iouv, maskv;
#pragma unroll
        for (int k = 0; k < 4; ++k) {
            const float ix1 = fmaxf(rb.x, cb[k].x);
            const float iy1 = fmaxf(rb.y, cb[k].y);
            const float ix2 = fminf(rb.z, cb[k].z);
            const float iy2 = fminf(rb.w, cb[k].w);
            const float inter = fmaxf(ix2 - ix1, 0.0f) * fmaxf(iy2 - iy1, 0.0f);
            const float uni   = rarea + carea[k] - inter;
            const float v     = inter / uni;
            iouv[k]  = v;
            maskv[k] = (v >= NB_IOU) ? 1.0f : 0.0f;
        }

        // streaming non-temporal 128-bit stores (coalesced across the wave)
        v4f* dst = (v4f*)(out_iou + (size_t)row * (size_t)ncols) + cg;
        __builtin_nontemporal_store(iouv, dst);
        if (WRITE_MASK) {
            v4f* mdst = (v4f*)(out_mask + (size_t)row * (size_t)ncols) + cg;
            __builtin_nontemporal_store(maskv, mdst);
        }
    }
}

// ---------------------------------------------------------------------------
// Launch: d_in[0] = detections (2000x4 f32), d_in[1] = gt_boxes (512x4 f32)
// d_out  = [ dt_gt_iou (2000*512) | dt_dt_iou (2000*2000) | mask (2000*2000) ]
// ---------------------------------------------------------------------------
extern "C" void kernel_launch(void* const* d_in, const int* in_sizes, int n_in,
                              void* d_out, int out_size, void* d_ws, size_t ws_size,
                              hipStream_t stream) {
    (void)in_sizes; (void)n_in; (void)out_size; (void)d_ws; (void)ws_size;

    const int N_DT = 2000;
    const int N_GT = 512;

    const v4f* dt = (const v4f*)d_in[0];
    const v4f* gt = (const v4f*)d_in[1];

    float* out    = (float*)d_out;
    float* dtgt   = out;                                   // 2000*512
    float* dtdt   = out + (size_t)N_DT * N_GT;             // 2000*2000
    float* nmask  = dtdt + (size_t)N_DT * N_DT;            // 2000*2000

    dim3 blk(256);

    // dt x gt : colGroups = 128 -> grid.x = 2 ; rowGroups = 500 -> grid.y = 125
    {
        dim3 grd((N_GT / 4 + 63) / 64, (N_DT / 4 + 3) / 4);
        iou_tile_kernel<false><<<grd, blk, 0, stream>>>(dt, N_DT, gt, N_GT, dtgt, nullptr);
    }
    // dt x dt : colGroups = 500 -> grid.x = 8 ; grid.y = 125  (fused IoU + mask)
    {
        dim3 grd((N_DT / 4 + 63) / 64, (N_DT / 4 + 3) / 4);
        iou_tile_kernel<true><<<grd, blk, 0, stream>>>(dt, N_DT, dt, N_DT, dtdt, nmask);
    }
}